// F_VAE_can_7902739824969
// MI455X (gfx1250) — compile-verified
//
#include <hip/hip_runtime.h>
#include <hip/hip_bf16.h>

// Problem constants (match reference)
#define NN   512     // latent dim
#define KSPA 64      // k_spa
#define ND   16      // n_domains
#define NB   1024    // batch
#define NSH  448     // NN - KSPA (shared prefix)
#define N_L  30688   // strictly-lower entries in last 64 rows: sum_{i=0}^{63}(448+i)

typedef __attribute__((ext_vector_type(2))) float v2f;
typedef __attribute__((ext_vector_type(8))) float v8f;

// ---------------------------------------------------------------------------
// Kernel 1: build F[d][i][c] for the last-64 rows of F_d = (I-L_d)^{-1} S_d.
// Recurrence per row i (i = row-448):
//   F[i][c] = base_i[c] + sum_{k<i} Llow[i][k] * F[k][c]
//   base_i[c] = L[448+i][c]               (c < 448)
//             = (c-448==i) ? S_emb[d][i]:0 (c >= 448)
// Each thread owns one column c -> its history lives in LDS, no barriers.
// Domain 0: L=0, S=I  ->  F row i is e_{448+i}.
// ---------------------------------------------------------------------------
__global__ __launch_bounds__(128) void build_F_kernel(
    const float* __restrict__ L_emb,   // [ND][N_L]
    const float* __restrict__ S_emb,   // [ND][KSPA]
    float* __restrict__ Fmat)          // [ND][64][512]
{
    __shared__ float hist[64][128];    // 32 KB: [row][local column]
    const int d  = blockIdx.x;         // 0..15
    const int c  = blockIdx.y * 128 + threadIdx.x;  // global column 0..511

    if (d == 0) {
        #pragma unroll 4
        for (int i = 0; i < 64; ++i)
            Fmat[(0 * 64 + i) * NN + c] = (c == NSH + i) ? 1.0f : 0.0f;
        return;
    }

    const float* Ld = L_emb + d * N_L;
    for (int i = 0; i < 64; ++i) {
        const int off = NSH * i + (i * (i - 1)) / 2;  // start of row 448+i in L_emb
        float acc;
        if (c < NSH) acc = Ld[off + c];
        else         acc = (c - NSH == i) ? S_emb[d * KSPA + i] : 0.0f;
        for (int k = 0; k < i; ++k) {
            const float w = Ld[off + NSH + k];        // Llow[i][k] (uniform across threads)
            acc += w * hist[k][threadIdx.x];
        }
        hist[i][threadIdx.x] = acc;
        Fmat[(d * 64 + i) * NN + c] = acc;
    }
}

// ---------------------------------------------------------------------------
// Kernel 2: out[b][0:448] = eps[b][0:448] + bias_shared   (float4-vectorized)
// ---------------------------------------------------------------------------
__global__ __launch_bounds__(256) void copy_bias_kernel(
    const float* __restrict__ eps,         // [NB][NN]
    const float* __restrict__ bias_shared, // [NSH]
    float* __restrict__ out)               // [NB][NN]
{
    const int idx = blockIdx.x * blockDim.x + threadIdx.x; // over NB*112 float4s
    const int b  = idx / (NSH / 4);
    const int c4 = idx % (NSH / 4);
    float4 e  = ((const float4*)(eps + b * NN))[c4];
    float4 bs = ((const float4*)bias_shared)[c4];
    float4 o  = make_float4(e.x + bs.x, e.y + bs.y, e.z + bs.z, e.w + bs.w);
    ((float4*)(out + b * NN))[c4] = o;
}

// ---------------------------------------------------------------------------
// Kernel 3: z[b][448+m] = F[d_b][m][:] . eps[b][:] + bias_nonshared[d_b][m]
// Grid: (batch-tile 0..63, row-tile 0..3, domain 0..15), one wave32 per block.
// 16x16 output tile accumulated with V_WMMA_F32_16X16X4_F32 over K=512.
// LDS staging (stride 20 dwords: 16B-aligned rows, conflict-free reads).
// Epilogue writes only columns whose batch element belongs to this domain.
// ---------------------------------------------------------------------------
__global__ __launch_bounds__(32) void spa_wmma_kernel(
    const float* __restrict__ eps,      // [NB][NN]
    const int*   __restrict__ didx,     // [NB]
    const float* __restrict__ Fmat,     // [ND][64][512]
    const float* __restrict__ bias_ns,  // [ND][KSPA]
    float* __restrict__ out)            // [NB][NN]
{
    const int bt   = blockIdx.x;        // batch tile (16 wide)
    const int rt   = blockIdx.y;        // output-row tile (16 of 64)
    const int dom  = blockIdx.z;        // domain
    const int lane = threadIdx.x;       // 0..31
    const int half = lane >> 4;         // 0|1
    const int l16  = lane & 15;

    __shared__ float As[16][20];        // F tile   [m][k], padded stride 20
    __shared__ float Bs[16][20];        // eps tile [n][k], padded stride 20

    const float* Fbase = Fmat + (dom * 64 + rt * 16) * NN;
    const float* Ebase = eps + (bt * 16) * NN;

    // cooperative tile-load mapping: lane -> (row, 8-float chunk)
    const int lrow = lane >> 1;         // 0..15
    const int lcol = (lane & 1) * 8;    // 0 | 8

    v8f acc = {0.f, 0.f, 0.f, 0.f, 0.f, 0.f, 0.f, 0.f};

    for (int kk = 0; kk < NN; kk += 16) {
        float4 fa0 = *(const float4*)(Fbase + lrow * NN + kk + lcol);
        float4 fa1 = *(const float4*)(Fbase + lrow * NN + kk + lcol + 4);
        float4 eb0 = *(const float4*)(Ebase + lrow * NN + kk + lcol);
        float4 eb1 = *(const float4*)(Ebase + lrow * NN + kk + lcol + 4);
        __syncthreads();                       // prior-iteration LDS reads done
        *(float4*)&As[lrow][lcol]     = fa0;
        *(float4*)&As[lrow][lcol + 4] = fa1;
        *(float4*)&Bs[lrow][lcol]     = eb0;
        *(float4*)&Bs[lrow][lcol + 4] = eb1;
        __syncthreads();                       // tiles visible

        #pragma unroll
        for (int ks = 0; ks < 16; ks += 4) {
            const int k0 = ks + half * 2;
            // A 16x4 f32: lane l -> M=l%16, VGPR0/1 = K = (l/16)*2 + {0,1}
            v2f a; a.x = As[l16][k0]; a.y = As[l16][k0 + 1];
            // B 4x16 f32: lane l -> N=l%16, VGPR0/1 = K = (l/16)*2 + {0,1}
            v2f b; b.x = Bs[l16][k0]; b.y = Bs[l16][k0 + 1];
            acc = __builtin_amdgcn_wmma_f32_16x16x4_f32(
                /*neg_a=*/false, a, /*neg_b=*/false, b,
                /*c_mod=*/(short)0, acc, /*reuse_a=*/false, /*reuse_b=*/false);
        }
    }

    // Epilogue (divergence only after all WMMAs; EXEC all-ones during matmul).
    // D layout: lane l, vgpr v -> M = v + (l/16)*8, N = l%16.
    const int b = bt * 16 + l16;
    if (didx[b] == dom) {
        #pragma unroll
        for (int v = 0; v < 8; ++v) {
            const int m = v + half * 8;
            out[b * NN + NSH + rt * 16 + m] =
                acc[v] + bias_ns[dom * KSPA + rt * 16 + m];
        }
    }
}

// ---------------------------------------------------------------------------
// Host launcher. Inputs (setup_inputs order):
//   0: epsilon        [NB*NN]  f32
//   1: d              [NB]     i32
//   2: L_emb          [ND*N_L] f32
//   3: S_emb          [ND*KSPA] f32
//   4: bias_nonshared [ND*KSPA] f32
//   5: bias_shared    [NSH]    f32
// Output: [NB*NN] f32.  Workspace: F matrices, ND*64*NN*4 = 2 MB.
// ---------------------------------------------------------------------------
extern "C" void kernel_launch(void* const* d_in, const int* in_sizes, int n_in,
                              void* d_out, int out_size, void* d_ws, size_t ws_size,
                              hipStream_t stream) {
    const float* eps     = (const float*)d_in[0];
    const int*   didx    = (const int*)  d_in[1];
    const float* L_emb   = (const float*)d_in[2];
    const float* S_emb   = (const float*)d_in[3];
    const float* bias_ns = (const float*)d_in[4];
    const float* bias_sh = (const float*)d_in[5];
    float*       out     = (float*)d_out;
    float*       Fmat    = (float*)d_ws;      // [ND][64][512]

    // 1) build per-domain F tail rows
    build_F_kernel<<<dim3(ND, NN / 128), dim3(128), 0, stream>>>(L_emb, S_emb, Fmat);

    // 2) shared prefix: copy + bias  (NB*NSH/4 float4s)
    const int n4 = NB * (NSH / 4);
    copy_bias_kernel<<<n4 / 256, 256, 0, stream>>>(eps, bias_sh, out);

    // 3) WMMA tail: 16 batch x 16 rows per wave, masked per-domain epilogue
    spa_wmma_kernel<<<dim3(NB / 16, KSPA / 16, ND), dim3(32), 0, stream>>>(
        eps, didx, Fmat, bias_ns, out);
}